// COTR_36386962932372
// MI455X (gfx1250) — compile-verified
//
#include <hip/hip_runtime.h>
#include <hip/hip_bf16.h>
#include <math.h>

// ---------------- problem constants ----------------
#define BS     1024
#define NB     6
#define NROW   (BS * NB)        // 6144 rows of `combined`
#define DCOMB  128              // 64 feat + 64 shape
#define NTILE  (NROW / 16)      // 384 16-row tiles
#define NGRP   (NTILE / 4)      // 96 groups of 4 col-tiles (64 cols)
#define NSUP   (NTILE / 8)      // 48 row super-tiles (128 rows)
#define BSTRIDE 132             // padded LDS row stride (floats): bank-conflict-free
#define EPSF   1e-8f

typedef float v2f __attribute__((ext_vector_type(2)));
typedef float v8f __attribute__((ext_vector_type(8)));

// =====================================================================
// Kernel 1: box features + MLP -> combined[6144][128], norms[6144]
// One 64-thread block per combined row (row = box*BS + sample).
// =====================================================================
__global__ void __launch_bounds__(64)
cotr_mlp_kernel(const float* __restrict__ feat, const float* __restrict__ bboxes,
                const float* __restrict__ w1,  const float* __restrict__ b1,
                const float* __restrict__ w2,  const float* __restrict__ b2,
                const float* __restrict__ wh1, const float* __restrict__ bh1,
                const float* __restrict__ wh2, const float* __restrict__ bh2,
                const float* __restrict__ wres,const float* __restrict__ bres,
                const float* __restrict__ wint,const float* __restrict__ bint,
                const float* __restrict__ wout,const float* __restrict__ bout,
                float* __restrict__ combined, float* __restrict__ norms,
                float* __restrict__ loss_out)
{
    const int row = blockIdx.x;         // 0..6143, box-major
    const int box = row / BS;
    const int smp = row % BS;
    const int tid = threadIdx.x;        // 0..63

    __shared__ float sx[6];
    __shared__ float sa[64];
    __shared__ float sb[64];
    __shared__ float sh[64];
    __shared__ float red[64];

    if (row == 0 && tid == 0) loss_out[0] = 0.0f;   // loss is identically zero

    if (tid == 0) {
        const float* bb = bboxes + (size_t)(smp * NB + box) * 4;
        float w = bb[2] - bb[0];
        float h = bb[3] - bb[1];
        sx[0] = w; sx[1] = h; sx[2] = w * h;
        sx[3] = w / h; sx[4] = w + h; sx[5] = sqrtf(w * w + h * h);
    }
    __syncthreads();

    if (tid < 32) {                       // 6 -> 32, ReLU
        float acc = b1[tid];
        #pragma unroll
        for (int i = 0; i < 6; ++i) acc += sx[i] * w1[i * 32 + tid];
        sa[tid] = fmaxf(acc, 0.0f);
    }
    __syncthreads();

    {                                     // 32 -> 64, ReLU  (h)
        float acc = b2[tid];
        #pragma unroll 8
        for (int i = 0; i < 32; ++i) acc += sa[i] * w2[i * 64 + tid];
        sh[tid] = fmaxf(acc, 0.0f);
    }
    __syncthreads();

    {                                     // hid1 = ReLU(h @ wh1 + bh1)
        float acc = bh1[tid];
        #pragma unroll 8
        for (int i = 0; i < 64; ++i) acc += sh[i] * wh1[i * 64 + tid];
        sa[tid] = fmaxf(acc, 0.0f);
    }
    __syncthreads();

    {                                     // hid2 = ReLU(hid1 @ wh2 + bh2)
        float acc = bh2[tid];
        #pragma unroll 8
        for (int i = 0; i < 64; ++i) acc += sa[i] * wh2[i * 64 + tid];
        sb[tid] = fmaxf(acc, 0.0f);
    }
    __syncthreads();

    {                                     // hid = hid2 + h @ wres + bres
        float acc = bres[tid] + sb[tid];
        #pragma unroll 8
        for (int i = 0; i < 64; ++i) acc += sh[i] * wres[i * 64 + tid];
        sa[tid] = acc;
    }
    __syncthreads();

    {                                     // inter = ReLU(hid @ wint + bint)
        float acc = bint[tid];
        #pragma unroll 8
        for (int i = 0; i < 64; ++i) acc += sa[i] * wint[i * 64 + tid];
        sb[tid] = fmaxf(acc, 0.0f);
    }
    __syncthreads();

    float outv = bout[tid];               // out = inter @ wout + bout
    #pragma unroll 8
    for (int i = 0; i < 64; ++i) outv += sb[i] * wout[i * 64 + tid];

    const float fv = feat[(size_t)(smp * NB + box) * 64 + tid];
    float* crow = combined + (size_t)row * DCOMB;
    crow[tid]      = fv;
    crow[64 + tid] = outv;

    red[tid] = fv * fv + outv * outv;
    __syncthreads();
    #pragma unroll
    for (int off = 32; off > 0; off >>= 1) {
        if (tid < off) red[tid] += red[tid + off];
        __syncthreads();
    }
    if (tid == 0) norms[row] = sqrtf(red[0]);
}

// =====================================================================
// Kernel 2: gram = combined @ combined^T via V_WMMA_F32_16X16X4_F32.
// Workgroup (8 waves) owns a 128x64 output region. The 64x128 B panel
// is staged once into LDS with GLOBAL_LOAD_ASYNC_TO_LDS_B128 (ASYNCcnt)
// and shared by all 8 waves; each wave computes a 16x64 strip from LDS.
// =====================================================================
__device__ __forceinline__ float cotr_fastdiv(float num, float den)
{
#if __has_builtin(__builtin_amdgcn_rcpf)
    return num * __builtin_amdgcn_rcpf(den);   // v_rcp_f32 + v_mul_f32
#else
    return num / den;
#endif
}

__device__ __forceinline__ void cotr_store_tile(v8f c, int ti, int tj, int lane,
                                                const float* __restrict__ norms,
                                                const float ni[8],
                                                float* __restrict__ sims)
{
    const int j  = tj * 16 + (lane & 15);
    const float nj = norms[j];
    #pragma unroll
    for (int v = 0; v < 8; ++v) {
        const int i = ti * 16 + v + ((lane >> 4) << 3);
        if (i < j) {
            const float cosv = cotr_fastdiv(c[v], fmaxf(ni[v] * nj, EPSF));
            const unsigned idx = (unsigned)i * (NROW - 1)
                               - (unsigned)((i * (i - 1)) >> 1)
                               + (unsigned)(j - i - 1);
            sims[idx] = cosv;
        }
    }
}

__global__ void __launch_bounds__(256)
cotr_gram_kernel(const float* __restrict__ combined,
                 const float* __restrict__ norms,
                 float* __restrict__ sims)
{
    const int g  = blockIdx.x;            // column group: cols [64g, 64g+64)
    const int rs = blockIdx.y;            // row super-tile: rows [128rs, 128rs+128)
    if (2 * rs > g) return;               // region entirely below diagonal

    // Sole __shared__ object in this kernel -> LDS offset 0.
    __shared__ float Bs[64 * BSTRIDE];    // 64 rows x 128 k, padded stride

    const int tid = threadIdx.x;          // 0..255

    // C-visible stores so the compiler can never conclude "Bs is never
    // written" and fold the LDS reads to undef (the real writes below are
    // done by the async DMA, invisible to the compiler).
    Bs[tid] = 0.0f;

    // ---- async-stage B panel: rows 64g..64g+63, 128 floats each ----
    // 2048 b128 chunks, 8 per thread, lane-adjacent chunks are 16B apart.
    // &Bs[0] is passed as a dummy operand so the LDS array escapes into the
    // "memory"-clobbering asm (forces the compiler to treat it as written).
    #pragma unroll
    for (int c = 0; c < 8; ++c) {
        const int chunk = tid + 256 * c;            // 0..2047
        const int row   = chunk >> 5;               // 0..63
        const int kq    = (chunk & 31) << 2;        // float offset 0..124
        const unsigned ldsoff = (unsigned)((row * BSTRIDE + kq) * 4);
        const float* gp = combined + (size_t)(g * 64 + row) * DCOMB + kq;
        asm volatile("global_load_async_to_lds_b128 %0, %1, off"
                     :: "v"(ldsoff), "v"(gp), "v"(&Bs[0]) : "memory");
    }
#if __has_builtin(__builtin_amdgcn_s_wait_asynccnt)
    __builtin_amdgcn_s_wait_asynccnt(0);
#else
    asm volatile("s_wait_asynccnt 0x0" ::: "memory");
#endif
    __syncthreads();

    const int wave = tid >> 5;
    const int lane = tid & 31;
    const int ti   = rs * 8 + wave;       // this wave's 16-row tile

    // Skip waves whose whole strip is below the diagonal (wave-uniform,
    // so EXEC stays all-1s inside the WMMA loop).
    if (16 * ti < 64 * g + 63) {
        const int l15 = lane & 15;
        const int kh  = (lane >> 4) << 1;           // 0 or 2

        const float* __restrict__ Ap = combined + (size_t)(ti * 16 + l15) * DCOMB + kh;
        const float* Bl = Bs + (size_t)l15 * BSTRIDE + kh;

        v8f c0 = {}, c1 = {}, c2 = {}, c3 = {};

        #pragma unroll 4
        for (int k = 0; k < DCOMB; k += 4) {
            const v2f a  = *(const v2f*)(Ap + k);                  // global_load_b64
            const v2f b0 = *(const v2f*)(Bl + k);                  // ds_load (2addr fused)
            const v2f b1 = *(const v2f*)(Bl + 16 * BSTRIDE + k);
            const v2f b2 = *(const v2f*)(Bl + 32 * BSTRIDE + k);
            const v2f b3 = *(const v2f*)(Bl + 48 * BSTRIDE + k);
            c0 = __builtin_amdgcn_wmma_f32_16x16x4_f32(false, a, false, b0, (short)0, c0, false, false);
            c1 = __builtin_amdgcn_wmma_f32_16x16x4_f32(false, a, false, b1, (short)0, c1, false, false);
            c2 = __builtin_amdgcn_wmma_f32_16x16x4_f32(false, a, false, b2, (short)0, c2, false, false);
            c3 = __builtin_amdgcn_wmma_f32_16x16x4_f32(false, a, false, b3, (short)0, c3, false, false);
        }

        float ni[8];
        #pragma unroll
        for (int v = 0; v < 8; ++v)
            ni[v] = norms[ti * 16 + v + ((lane >> 4) << 3)];

        cotr_store_tile(c0, ti, 4 * g + 0, lane, norms, ni, sims);
        cotr_store_tile(c1, ti, 4 * g + 1, lane, norms, ni, sims);
        cotr_store_tile(c2, ti, 4 * g + 2, lane, norms, ni, sims);
        cotr_store_tile(c3, ti, 4 * g + 3, lane, norms, ni, sims);
    }
}

// =====================================================================
extern "C" void kernel_launch(void* const* d_in, const int* in_sizes, int n_in,
                              void* d_out, int out_size, void* d_ws, size_t ws_size,
                              hipStream_t stream) {
    const float* feat   = (const float*)d_in[0];
    const float* bboxes = (const float*)d_in[1];
    const float* w1     = (const float*)d_in[2];
    const float* b1     = (const float*)d_in[3];
    const float* w2     = (const float*)d_in[4];
    const float* b2     = (const float*)d_in[5];
    const float* wh1    = (const float*)d_in[6];
    const float* bh1    = (const float*)d_in[7];
    const float* wh2    = (const float*)d_in[8];
    const float* bh2    = (const float*)d_in[9];
    const float* wres   = (const float*)d_in[10];
    const float* bres   = (const float*)d_in[11];
    const float* wint   = (const float*)d_in[12];
    const float* bint   = (const float*)d_in[13];
    const float* wout   = (const float*)d_in[14];
    const float* bout   = (const float*)d_in[15];

    float* combined = (float*)d_ws;                       // 6144*128 floats = 3 MB
    float* norms    = combined + (size_t)NROW * DCOMB;    // 6144 floats
    float* loss     = (float*)d_out;                      // d_out[0]
    float* sims     = (float*)d_out + 1;                  // 18,871,296 floats

    cotr_mlp_kernel<<<NROW, 64, 0, stream>>>(
        feat, bboxes, w1, b1, w2, b2, wh1, bh1, wh2, bh2,
        wres, bres, wint, bint, wout, bout, combined, norms, loss);

    cotr_gram_kernel<<<dim3(NGRP, NSUP), 256, 0, stream>>>(combined, norms, sims);
}